// Minimax_Conv2D_90855738180337
// MI455X (gfx1250) — compile-verified
//
#include <hip/hip_runtime.h>

typedef unsigned int v4u __attribute__((ext_vector_type(4)));
typedef int          v8i __attribute__((ext_vector_type(8)));
typedef int          v4i __attribute__((ext_vector_type(4)));

#define NB 16
#define NC 64
#define NH 64
#define NW 64
#define NO 128

// One block per (b, h): stage x[b, :, h-1..h+1, :] (replicate-clamped rows)
// into LDS with one TDM tensor_load_to_lds, precompute per-(o,tap) metadata
// from conn into LDS, then 256 threads produce all 128 output channels for
// the 64 pixels of this row.
__global__ __launch_bounds__(256) void minimax_conv2d_kernel(
    const float* __restrict__ x,    // [B, C, H, W]
    const float* __restrict__ w1,   // [O, 9]
    const float* __restrict__ w2,   // [O, 3]
    const int*   __restrict__ conn, // [O*9], values in [0, C*9)
    float*       __restrict__ out)  // [B, O, H, W]
{
    __shared__ float    tile[NC * 3 * NW];  // 48 KB: [c][slot][w], slot-major per channel
    __shared__ unsigned meta[NO * 9];       // packed: (ldsElemBase << 2) | j
    __shared__ float    tw1[NO * 9];
    __shared__ float    tw2[NO * 3];

    const int tid = (int)threadIdx.x;
    const int bh  = (int)blockIdx.x;
    const int b   = bh >> 6;
    const int h   = bh & 63;

    const int r0    = (h == 0) ? 0 : (h - 1);
    const int r1    = (h == NH - 1) ? (NH - 1) : (h + 1);
    const int nrows = r1 - r0 + 1;          // 2 at top/bottom boundary, else 3

    // ---- TDM: issue one 3D-tile load (W x nrows x C) from wave 0 only ----
    if (tid == 0) {
        unsigned long long gaddr =
            (unsigned long long)(size_t)(x + ((size_t)b * NC * NH + (size_t)r0) * NW);
        unsigned ldsa = (unsigned)(unsigned long long)(size_t)&tile[0];

        // D# group 0: count=1 | lds_addr | global_addr[56:0] | type=2
        v4u g0 = { 1u,
                   ldsa,
                   (unsigned)(gaddr & 0xFFFFFFFFull),
                   (unsigned)((gaddr >> 32) & 0x01FFFFFFull) | (2u << 30) };
        // D# group 1:
        //  d0: wg_mask=0, data_size=2 (4B)
        //  d1: abar_addr=0 | tensor_dim0.lo16 = 64
        //  d2: tensor_dim0.hi16=0 | tensor_dim1.lo16 = 64
        //  d3: tensor_dim1.hi16=0 | tile_dim0 = 64
        //  d4: tile_dim1 = nrows | tile_dim2 = 64
        //  d5: tensor_dim0_stride.lo32 = 64 (elements between rows)
        //  d6: dim0_stride.hi16=0 | tensor_dim1_stride.lo16 = 4096 (elements between channels)
        //  d7: tensor_dim1_stride.hi32 = 0
        v8i g1 = { (int)(2u << 16),
                   (int)(64u << 16),
                   (int)(64u << 16),
                   (int)(64u << 16),
                   (int)((64u << 16) | (unsigned)nrows),
                   64,
                   (int)(0x1000u << 16),
                   0 };
        // D# group 2: tensor_dim2 = 64; dim3/dim2_stride/tile_dim3 unused
        v4i g2 = { 64, 0, 0, 0 };
        v4i g3 = { 0, 0, 0, 0 };
        // Extra 8-dword operand required by the 6-arg builtin variant (zero-filled)
        v8i g4 = { 0, 0, 0, 0, 0, 0, 0, 0 };

        __builtin_amdgcn_tensor_load_to_lds(g0, g1, g2, g3, g4, /*cpol=*/0);
        __builtin_amdgcn_s_wait_tensorcnt(0);
    }

    // ---- per-(o,tap) metadata: decode conn once per block (overlaps TDM) ----
    for (int t = tid; t < NO * 9; t += 256) {
        int idx = conn[t];
        int c   = idx / 9;
        int k   = idx - c * 9;
        int i   = k / 3;
        int jj  = k - i * 3;
        int row = h - 1 + i;
        row = row < 0 ? 0 : (row > NH - 1 ? NH - 1 : row);
        int slot = row - r0;                     // clamped row -> LDS slot
        meta[t] = ((unsigned)((c * nrows + slot) * NW) << 2) | (unsigned)jj;
        tw1[t]  = w1[t];
    }
    for (int t = tid; t < NO * 3; t += 256) tw2[t] = w2[t];
    __syncthreads();

    // ---- compute: lane = pixel w, wave-uniform channel group ----
    const int w = tid & 63;
    const int g = tid >> 6;                      // 4 groups x 32 channels
    const int c0 = (w == 0) ? 0 : (w - 1);       // clamped column offsets (elements)
    const int c1 = w;
    const int c2 = (w == NW - 1) ? (NW - 1) : (w + 1);

    float* outp = out + (size_t)b * NO * NH * NW + (size_t)h * NW + w;

    for (int oi = 0; oi < 32; ++oi) {
        const int o = (g << 5) | oi;
        const unsigned* mp  = &meta[o * 9];
        const float*    w1p = &tw1[o * 9];
        float res = 3.402823466e38f;
        #pragma unroll
        for (int rr = 0; rr < 3; ++rr) {
            float ma = 0.0f;                     // |.| >= 0, so 0 is a safe identity
            #pragma unroll
            for (int t = 0; t < 3; ++t) {
                unsigned m = mp[rr * 3 + t];
                int jj   = (int)(m & 3u);
                int base = (int)(m >> 2);
                int col  = (jj == 0) ? c0 : ((jj == 1) ? c1 : c2);
                float v  = tile[base + col];
                float d  = fabsf(v - w1p[rr * 3 + t]);
                ma = fmaxf(ma, d);
            }
            res = fminf(res, fabsf(ma - tw2[o * 3 + rr]));
        }
        outp[(size_t)o * (NH * NW)] = res;
    }
}

extern "C" void kernel_launch(void* const* d_in, const int* in_sizes, int n_in,
                              void* d_out, int out_size, void* d_ws, size_t ws_size,
                              hipStream_t stream) {
    (void)in_sizes; (void)n_in; (void)out_size; (void)d_ws; (void)ws_size;
    const float* x    = (const float*)d_in[0];
    const float* w1   = (const float*)d_in[1];
    const float* w2   = (const float*)d_in[2];
    const int*   conn = (const int*)d_in[3];
    float*       out  = (float*)d_out;

    dim3 grid(NB * NH);   // 1024 blocks: one per (batch, output row)
    dim3 block(256);      // 8 wave32
    minimax_conv2d_kernel<<<grid, block, 0, stream>>>(x, w1, w2, conn, out);
}